// SpiralConvConvBlock_33028298506565
// MI455X (gfx1250) — compile-verified
//
#include <hip/hip_runtime.h>
#include <stdint.h>

// Problem shape (fixed by reference setup_inputs)
#define B_ 4
#define L_ 4096
#define D_ 1024
#define C_ 64               // chunks along L
#define LC_ (L_ / C_)       // 64 timesteps per chunk
#define GROUP_ 16           // rows staged per async group
#define NG_ (LC_ / GROUP_)  // 4 groups per chunk
#define BLOCK_D_ 256        // threads per block = channels per block (8 waves)
#define NWAVES_ (BLOCK_D_ / 32)

// ---- CDNA5 async global->LDS + split wait counters (inline asm) ----
__device__ __forceinline__ void async_load_b128(uint32_t lds_addr, uint64_t gaddr) {
  asm volatile("global_load_async_to_lds_b128 %0, %1, off"
               :: "v"(lds_addr), "v"(gaddr) : "memory");
}
__device__ __forceinline__ void wait_async_le4() {
  asm volatile("s_wait_asynccnt 4" ::: "memory");
}
__device__ __forceinline__ void wait_async_0() {
  asm volatile("s_wait_asynccnt 0" ::: "memory");
}
__device__ __forceinline__ void wait_ds_0() {
  asm volatile("s_wait_dscnt 0" ::: "memory");
}

// Passes 1 & 3: chunked scan u = a*u + h*x over LC_ timesteps.
// PASS3=false: start from 0, emit chunk-final (complex) to chunk_final.
// PASS3=true : start from carry_in, emit Re(u) per timestep to out.
template <bool PASS3>
__global__ __launch_bounds__(BLOCK_D_)
void scan_chunk_kernel(const float* __restrict__ x,
                       const float* __restrict__ phazor,
                       const float* __restrict__ phazor_init,
                       const float2* __restrict__ carry_in,
                       float* __restrict__ out,
                       float2* __restrict__ chunk_final)
{
  // per-wave double-buffered tile: [wave][buf][16 rows x 32 channels]
  __shared__ float tile[NWAVES_][2][GROUP_ * 32];

  const int tid   = threadIdx.x;
  const int wave  = tid >> 5;
  const int lane  = tid & 31;
  const int d     = blockIdx.x * BLOCK_D_ + tid;   // this thread's channel
  const int c     = blockIdx.y;                    // chunk index
  const int b     = blockIdx.z;                    // batch
  const int t0    = c * LC_;
  const int db    = blockIdx.x * BLOCK_D_ + (wave << 5); // wave's base channel
  const int laneR = lane >> 3;        // 0..3 : row within a b128 instruction
  const int laneD = (lane & 7) << 2;  // 0,4,..,28 : channel offset (4 floats)

  const uint64_t rowBytes = 4ull * D_;
  // lane's global byte address for (row = t0+laneR, channel = db+laneD)
  uint64_t gbase = (uint64_t)(uintptr_t)x +
                   4ull * ((uint64_t)(b * L_ + t0 + laneR) * D_ + db + laneD);
  const uint32_t ldsoff = (uint32_t)((laneR * 32 + laneD) << 2);
  const uint32_t lds0 = (uint32_t)(uintptr_t)(&tile[wave][0][0]) + ldsoff;
  const uint32_t lds1 = (uint32_t)(uintptr_t)(&tile[wave][1][0]) + ldsoff;

  // Kick off group 0 into buffer 0 before doing any math.
#pragma unroll
  for (int i = 0; i < 4; ++i)
    async_load_b128(lds0 + (uint32_t)(i * 512),
                    gbase + (uint64_t)(i * 4) * rowBytes);

  // Per-channel parameters: a = ph/|ph| * exp(-|ph|), h = ph_init
  const float pr  = phazor[2 * d];
  const float pim = phazor[2 * d + 1];
  const float mag = sqrtf(pr * pr + pim * pim);
  const float sc  = expf(-mag) / mag;
  const float ar = pr * sc, ai = pim * sc;
  const float hr = phazor_init[2 * d];
  const float hi = phazor_init[2 * d + 1];

  float ur, ui;
  if (PASS3) {
    const float2 cin = carry_in[(size_t)(b * C_ + c) * D_ + d];
    ur = cin.x; ui = cin.y;
  } else {
    ur = 0.0f; ui = 0.0f;
  }

  for (int g = 0; g < NG_; ++g) {
    if (g + 1 < NG_) {
      const uint32_t ldsn = ((g + 1) & 1) ? lds1 : lds0;
      const uint64_t gg   = gbase + (uint64_t)((g + 1) * GROUP_) * rowBytes;
      if (g >= 1) wait_ds_0();  // our reads of the recycled buffer are done
#pragma unroll
      for (int i = 0; i < 4; ++i)
        async_load_b128(ldsn + (uint32_t)(i * 512),
                        gg + (uint64_t)(i * 4) * rowBytes);
      wait_async_le4();         // in-order completion => current group landed
    } else {
      wait_async_0();
    }
    const float* tb = &tile[wave][g & 1][0];
#pragma unroll
    for (int r = 0; r < GROUP_; ++r) {
      const float xv = tb[r * 32 + lane];
      const float nr = fmaf(ar, ur, fmaf(-ai, ui, hr * xv));
      const float ni = fmaf(ai, ur, fmaf(ar, ui, hi * xv));
      ur = nr; ui = ni;
      if (PASS3)
        out[(size_t)(b * L_ + t0 + g * GROUP_ + r) * D_ + d] = ur;
    }
  }

  if (!PASS3)
    chunk_final[(size_t)(b * C_ + c) * D_ + d] = make_float2(ur, ui);
}

// Pass 2: scan the C_ chunk-finals per (b,d) with factor a^LC_,
// starting from last_conv; record carry-in per chunk.
__global__ __launch_bounds__(256)
void carry_scan_kernel(const float* __restrict__ phazor,
                       const float* __restrict__ last_re,
                       const float* __restrict__ last_im,
                       const float2* __restrict__ S,
                       float2* __restrict__ In)
{
  const int tid = blockIdx.x * 256 + threadIdx.x; // 0 .. B_*D_-1
  const int b = tid / D_;
  const int d = tid % D_;

  const float pr  = phazor[2 * d];
  const float pim = phazor[2 * d + 1];
  const float mag = sqrtf(pr * pr + pim * pim);
  const float sc  = expf(-mag) / mag;
  float cr = pr * sc, ci = pim * sc;
  // a^64 via 6 complex squarings
#pragma unroll
  for (int i = 0; i < 6; ++i) {
    const float nr = cr * cr - ci * ci;
    const float ni = 2.0f * cr * ci;
    cr = nr; ci = ni;
  }

  float ur = last_re[d];
  float ui = last_im[d];
#pragma unroll 4
  for (int c = 0; c < C_; ++c) {
    const size_t idx = (size_t)(b * C_ + c) * D_ + d;
    In[idx] = make_float2(ur, ui);
    const float2 s = S[idx];
    const float nr = fmaf(cr, ur, fmaf(-ci, ui, s.x));
    const float ni = fmaf(ci, ur, fmaf(cr, ui, s.y));
    ur = nr; ui = ni;
  }
}

extern "C" void kernel_launch(void* const* d_in, const int* in_sizes, int n_in,
                              void* d_out, int out_size, void* d_ws, size_t ws_size,
                              hipStream_t stream) {
  const float* x           = (const float*)d_in[0];
  const float* phazor      = (const float*)d_in[1];
  const float* phazor_init = (const float*)d_in[2];
  const float* last_re     = (const float*)d_in[3];
  const float* last_im     = (const float*)d_in[4];
  float* out = (float*)d_out;

  float2* S  = (float2*)d_ws;                       // B*C*D complex (2 MB)
  float2* In = S + (size_t)B_ * C_ * D_;            // B*C*D complex (2 MB)

  dim3 grid(D_ / BLOCK_D_, C_, B_);                 // (4, 64, 4)

  scan_chunk_kernel<false><<<grid, BLOCK_D_, 0, stream>>>(
      x, phazor, phazor_init, nullptr, nullptr, S);
  carry_scan_kernel<<<(B_ * D_) / 256, 256, 0, stream>>>(
      phazor, last_re, last_im, S, In);
  scan_chunk_kernel<true><<<grid, BLOCK_D_, 0, stream>>>(
      x, phazor, phazor_init, In, out, nullptr);
}